// EELayer_65859028517066
// MI455X (gfx1250) — compile-verified
//
#include <hip/hip_runtime.h>
#include <hip/hip_bf16.h>

typedef float v2f __attribute__((ext_vector_type(2)));
typedef float v8f __attribute__((ext_vector_type(8)));

// ---------------------------------------------------------------------------
// C[m, o] = sum_k A[m,k] * W[o,k]     (A: [M,K] row-major, W: [O,K] row-major)
// One wave32 per 16x16 output tile, fp32 WMMA 16x16x4.
// blockDim.x = 32 * (O/16): wave w in the block handles column tile w.
// ---------------------------------------------------------------------------
__global__ void gemm_nt_wmma(const float* __restrict__ A,
                             const float* __restrict__ W,
                             float* __restrict__ C,
                             int M, int K, int O) {
  const int lane  = threadIdx.x & 31;
  const int oTile = threadIdx.x >> 5;
  const int mTile = blockIdx.x;
  if (mTile * 16 >= M) return;                  // uniform per wave -> EXEC all-1s
  const int half = lane >> 4;                   // 0: lanes 0-15, 1: lanes 16-31
  const int l16  = lane & 15;

  // A frag: lane(l16,half) VGPR0 = A[m=l16][k+2*half], VGPR1 = A[m=l16][k+2*half+1]
  const float* aPtr = A + (size_t)(mTile * 16 + l16) * K + 2 * half;
  // B frag (B = W^T): lane holds column o=l16, K split identically to A
  const float* wPtr = W + (size_t)(oTile * 16 + l16) * K + 2 * half;

  v8f acc = {};
#pragma unroll 8
  for (int k = 0; k < K; k += 4) {
    v2f a = *(const v2f*)(aPtr + k);
    v2f b = *(const v2f*)(wPtr + k);
    acc = __builtin_amdgcn_wmma_f32_16x16x4_f32(
        /*neg_a=*/false, a, /*neg_b=*/false, b,
        /*c_mod=*/(short)0, acc, /*reuse_a=*/false, /*reuse_b=*/false);
  }

  // D layout: VGPR r -> row (half*8 + r), lane%16 -> column
  float* cPtr = C + (size_t)(mTile * 16 + half * 8) * O + oTile * 16 + l16;
#pragma unroll
  for (int r = 0; r < 8; ++r) cPtr[(size_t)r * O] = acc[r];
}

// ---------------------------------------------------------------------------
// Pass A: per-edge logit e = dot(z[src], df[dst]) + atomic segment-max keys.
// One wave32 per edge; lane loads float2 of the 64-float rows (coalesced 256B).
// ---------------------------------------------------------------------------
__device__ __forceinline__ unsigned float_key(float f) {
  unsigned u = __float_as_uint(f);
  return (u & 0x80000000u) ? ~u : (u | 0x80000000u);   // monotone float->u32
}
__device__ __forceinline__ float key_float(unsigned k) {
  unsigned u = (k & 0x80000000u) ? (k & 0x7FFFFFFFu) : ~k;
  return __uint_as_float(u);
}

__global__ void edge_logits(const float* __restrict__ z,
                            const float* __restrict__ df,
                            const int* __restrict__ src,
                            const int* __restrict__ dst,
                            float* __restrict__ e,
                            unsigned* __restrict__ emaxKey, int E) {
  const int edge = (int)((blockIdx.x * (size_t)blockDim.x + threadIdx.x) >> 5);
  const int lane = threadIdx.x & 31;
  if (edge >= E) return;                        // whole wave exits together
  const int s = src[edge], d = dst[edge];
  v2f a = *(const v2f*)(z  + (size_t)s * 64 + lane * 2);
  v2f b = *(const v2f*)(df + (size_t)d * 64 + lane * 2);
  float p = a.x * b.x + a.y * b.y;
#pragma unroll
  for (int off = 16; off; off >>= 1) p += __shfl_xor(p, off, 32);
  if (lane == 0) {
    e[edge] = p;
    atomicMax(emaxKey + d, float_key(p));
  }
}

// ---------------------------------------------------------------------------
// Pass B: ex = exp(e - emax[dst]); num[dst] += ex * z[src]; denom[dst] += ex.
// One wave32 per edge, 2 f32 atomic adds per lane (global_atomic_add_f32,
// L2-resident RMW).
// ---------------------------------------------------------------------------
__global__ void edge_scatter(const float* __restrict__ z,
                             const float* __restrict__ e,
                             const unsigned* __restrict__ emaxKey,
                             const int* __restrict__ src,
                             const int* __restrict__ dst,
                             float* __restrict__ num,
                             float* __restrict__ denom, int E) {
  const int edge = (int)((blockIdx.x * (size_t)blockDim.x + threadIdx.x) >> 5);
  const int lane = threadIdx.x & 31;
  if (edge >= E) return;
  const int s = src[edge], d = dst[edge];
  const float m  = key_float(emaxKey[d]);       // only read for dsts with edges
  const float ex = __expf(e[edge] - m);
  v2f a = *(const v2f*)(z + (size_t)s * 64 + lane * 2);
  float* np = num + (size_t)d * 64 + lane * 2;
  atomicAdd(np + 0, ex * a.x);
  atomicAdd(np + 1, ex * a.y);
  if (lane == 0) atomicAdd(denom + d, ex);
}

// ---------------------------------------------------------------------------
// Pass C: out = num / max(denom, !=0)
// ---------------------------------------------------------------------------
__global__ void finalize(const float* __restrict__ num,
                         const float* __restrict__ denom,
                         float* __restrict__ out, int total) {
  const int i = blockIdx.x * blockDim.x + threadIdx.x;
  if (i >= total) return;
  const float dn = denom[i >> 6];               // 64 outputs per node
  out[i] = num[i] / (dn == 0.0f ? 1.0f : dn);
}

// ---------------------------------------------------------------------------
extern "C" void kernel_launch(void* const* d_in, const int* in_sizes, int n_in,
                              void* d_out, int out_size, void* d_ws, size_t ws_size,
                              hipStream_t stream) {
  const float* h     = (const float*)d_in[0];   // [N, 256]
  const float* feat  = (const float*)d_in[1];   // [N, 64]
  const float* W_fc  = (const float*)d_in[2];   // [64, 256]
  const float* W_dst = (const float*)d_in[3];   // [64, 64]
  const int*   src   = (const int*)d_in[4];     // [E]
  const int*   dst   = (const int*)d_in[5];     // [E]
  float*       out   = (float*)d_out;           // [N, 64]

  const int FEAT = 64, O = 64;
  const int N   = in_sizes[1] / FEAT;
  const int Kfc = in_sizes[0] / N;
  const int E   = in_sizes[4];

  // Scratch layout (~84 MB): z | df | num | denom | emaxKey | e
  float*    z       = (float*)d_ws;
  float*    df      = z  + (size_t)N * O;
  float*    num     = df + (size_t)N * O;
  float*    denom   = num + (size_t)N * O;
  unsigned* emaxKey = (unsigned*)(denom + N);
  float*    e       = (float*)(emaxKey + N);

  // Zero the accumulators (num, denom) and max-keys (key 0 == ~ -inf sentinel).
  hipMemsetAsync(num, 0, ((size_t)N * O + 2 * (size_t)N) * sizeof(float), stream);

  // Node projections via fp32 WMMA.
  const dim3 gemmBlk(32 * (O / 16));            // 4 waves: one per 16-col tile
  const int  mTiles = (N + 15) / 16;
  gemm_nt_wmma<<<mTiles, gemmBlk, 0, stream>>>(h,    W_fc,  z,  N, Kfc,  O);
  gemm_nt_wmma<<<mTiles, gemmBlk, 0, stream>>>(feat, W_dst, df, N, FEAT, O);

  // Edge phase: 8 waves (8 edges) per 256-thread block.
  const int blkE = 256, edgesPerBlk = blkE / 32;
  const int gE = (E + edgesPerBlk - 1) / edgesPerBlk;
  edge_logits <<<gE, blkE, 0, stream>>>(z, df, src, dst, e, emaxKey, E);
  edge_scatter<<<gE, blkE, 0, stream>>>(z, e, emaxKey, src, dst, num, denom, E);

  const int total = N * O;
  finalize<<<(total + 255) / 256, 256, 0, stream>>>(num, denom, out, total);
}